// UserEmbedding_61873298866785
// MI455X (gfx1250) — compile-verified
//
#include <hip/hip_runtime.h>

// out[b, d] = W[d * num_users + x[b]]   (pure gather, 0 FLOPs)
//
// Traffic: ~1M independent 4B loads, each on its own 128B line (~134 MB of
// cacheline traffic, fits in 192 MB L2; ~22% inter-batch line reuse -> keep
// default RT policy on loads). Output is 4 MiB write-once -> NT stores.
// Bound by outstanding-load count, so: 8 independent gathers per thread
// issued back-to-back, coalesced b128 NT stores, 4096 wave32 waves.

#define DIM 64

// Clang-native 128-bit vector for __builtin_nontemporal_store
typedef float v4f __attribute__((ext_vector_type(4)));

__global__ __launch_bounds__(256) void user_embedding_gather(
    const int*  __restrict__ x,     // [batch]
    const float* __restrict__ W,    // [DIM, num_users] row-major
    float*      __restrict__ out,   // [batch, DIM]
    int batch,
    long long   num_users)
{
    // thread t -> batch element b = t/8, dims d0..d0+7 with d0 = (t%8)*8
    int t  = blockIdx.x * blockDim.x + threadIdx.x;
    int b  = t >> 3;
    if (b >= batch) return;
    int d0 = (t & 7) << 3;

    // x[b] is uniform across the 8 lanes covering this b -> broadcast, cache hit.
    long long col = (long long)x[b];

    // Base address of (d0, col); successive dims are num_users floats apart
    // (4 MB stride -> each load is an independent cacheline gather).
    const float* base = W + (long long)d0 * num_users + col;

    // 8 independent loads; compiler emits the global_load_b32 burst before
    // any s_wait_loadcnt, giving 8 gathers in flight per thread.
    float r0 = base[0LL * num_users];
    float r1 = base[1LL * num_users];
    float r2 = base[2LL * num_users];
    float r3 = base[3LL * num_users];
    float r4 = base[4LL * num_users];
    float r5 = base[5LL * num_users];
    float r6 = base[6LL * num_users];
    float r7 = base[7LL * num_users];

    // Stores: lane-consecutive d -> fully coalesced b128 stores.
    // Output is written once and never re-read by this dispatch: use
    // non-temporal stores (TH=NT) so the 4 MiB result doesn't evict the
    // gather working set from L2.
    v4f lo = { r0, r1, r2, r3 };
    v4f hi = { r4, r5, r6, r7 };
    v4f* dst = (v4f*)(out + (size_t)b * DIM + d0);
    __builtin_nontemporal_store(lo, dst);
    __builtin_nontemporal_store(hi, dst + 1);
}

extern "C" void kernel_launch(void* const* d_in, const int* in_sizes, int n_in,
                              void* d_out, int out_size, void* d_ws, size_t ws_size,
                              hipStream_t stream) {
    const int*   x = (const int*)d_in[0];      // [batch] int32
    const float* W = (const float*)d_in[1];    // [DIM * num_users] fp32

    int       batch     = in_sizes[0];
    long long num_users = (long long)in_sizes[1] / DIM;

    float* out = (float*)d_out;

    int threads = batch * 8;                   // 8 dims per thread
    int block   = 256;                         // 8 wave32 waves per block
    int grid    = (threads + block - 1) / block;

    user_embedding_gather<<<grid, block, 0, stream>>>(x, W, out, batch, num_users);
}